// NFS_57698590655197
// MI455X (gfx1250) — compile-verified
//
#include <hip/hip_runtime.h>
#include <hip/hip_bf16.h>

typedef __attribute__((ext_vector_type(2))) float v2f;
typedef __attribute__((ext_vector_type(8))) float v8f;

// ---------------------------------------------------------------------------
// Generic strided f32 WMMA: C(16x16) += A(16xK) * B(Kx16)
//   A element (m,k) at A[m*Am + k*Ak],  B element (k,n) at B[k*Bk + n*Bn].
// f32 16x16x4 VGPR layout: lanes 0-15 carry (row r, K0/K1), lanes 16-31 (row r, K2/K3).
// ---------------------------------------------------------------------------
template<int K>
__device__ inline v8f wmma_gen(const float* __restrict__ A, int Am, int Ak,
                               const float* __restrict__ B, int Bk, int Bn, v8f acc) {
  const int lane = threadIdx.x & 31;
  const int r    = lane & 15;
  const int koff = (lane >> 4) << 1;   // 0 or 2
#pragma unroll
  for (int k0 = 0; k0 < K; k0 += 4) {
    const int ka = k0 + koff, kb = ka + 1;
    v2f a, b;
    a.x = A[r * Am + ka * Ak];
    a.y = A[r * Am + kb * Ak];
    b.x = B[ka * Bk + r * Bn];
    b.y = B[kb * Bk + r * Bn];
    acc = __builtin_amdgcn_wmma_f32_16x16x4_f32(false, a, false, b, (short)0, acc,
                                                false, false);
  }
  return acc;
}

// C/D 16x16 f32 layout: VGPR i, lanes 0-15 -> (M=i, N=lane); lanes 16-31 -> (M=i+8).
__device__ inline void tile_store(float* __restrict__ C, int ldc, v8f acc, float scale,
                                  const float* colb, const float* rowb,
                                  bool do_gelu, bool accum) {
  const int lane = threadIdx.x & 31;
  const int col  = lane & 15;
  const int rb   = (lane >> 4) << 3;   // 0 or 8
#pragma unroll
  for (int i2 = 0; i2 < 8; ++i2) {
    float v = acc[i2] * scale;
    if (colb) v += colb[col];
    if (rowb) v += rowb[rb + i2];
    if (do_gelu) v = 0.5f * v * (1.0f + erff(v * 0.70710678118654752f));
    float* p = &C[(rb + i2) * ldc + col];
    if (accum) *p += v; else *p = v;
  }
}

struct GmlpP { const float *f1_b, *f1_w, *f2_b, *f2_w, *ln_b, *ln_g, *proj_b, *proj_w; };
struct EstP {
  const float *fc1_w, *fc2_w, *fc3_w;
  GmlpP gl[3];
  const float *k_b, *k_w, *o_b, *o_w, *q_b, *q_w;
  const float *se1_b, *se1_w, *se2_b, *se2_w, *se_conv_b, *se_conv_w;
  const float *v_b, *v_w;
};

__device__ inline float softplusf(float x) {
  return (x > 20.f) ? x : log1pf(__expf(x));
}

// ---------------------------------------------------------------------------
// Kernel A: estimator (both heads) + ang/mag + channel-combined complex filter
// One wave (32 threads) per batch element. hs==0: shift head, hs==1: scale head.
// ---------------------------------------------------------------------------
__global__ __launch_bounds__(32) void NFS_est_kernel(
    const float* __restrict__ c, const float* __restrict__ g,
    EstP Psh, EstP Psc, const float* __restrict__ projw,
    float* __restrict__ outMag, float* __restrict__ outAng,
    float* __restrict__ zws) {
  __shared__ float s_h[16 * 128];   // 2048
  __shared__ float s_big[8192];     // attention scratch | gmlp z(64x128)
  __shared__ float s_x3[4096];      // at2 | hn | vp/uv | head-tail t
  __shared__ float s_stat[32];

  const int n    = blockIdx.x;
  const int lane = threadIdx.x;
  const float* cb = c + (size_t)n * 1024;
  const float gval = g[n];
  float rpin[16];
  float zacc_r[17], zacc_i[17];

  float* s_qq = s_big;              // 16x64
  float* s_kk = s_big + 1024;       // 16x64
  float* s_vv = s_big + 2048;       // 16x64
  float* s_q  = s_big + 3072;       // 16x128
  float* s_cc = s_big + 5120;       // 16x128
  float* s_aw = s_big + 3072;       // 64x64 (overwrites q/cc after use)
  float* s_at = s_big + 7168;       // 16x64
  float* s_z  = s_big;              // 64x128 (gmlp phase)
  float* s_at2 = s_x3;              // 16x128
  float* s_hn  = s_x3;              // 16x128
  float* s_vp  = s_x3;              // 32x128
  float* s_t   = s_x3;              // 16x128

  for (int hs = 0; hs < 2; ++hs) {
    const EstP& P = (hs == 0) ? Psh : Psc;

    // --- pin (kept in regs), q = fc1@pin, cc = fc2@csl ---
#pragma unroll
    for (int t = 0; t < 16; ++t) rpin[t] = cb[lane + 32 * t];
    for (int idx = lane; idx < 2048; idx += 32) {
      int o = idx >> 7, w = idx & 127;
      float a1 = 0.f, a2 = 0.f;
#pragma unroll
      for (int k = 0; k < 4; ++k) a1 += P.fc1_w[o * 4 + k] * cb[k * 128 + w];
#pragma unroll
      for (int k = 0; k < 3; ++k) a2 += P.fc2_w[o * 3 + k] * cb[(3 + k) * 128 + w];
      s_q[idx] = a1;
      s_cc[idx] = a2;
    }
    __syncthreads();

    // --- qq/kk/vv = act(16x128) @ W^T(128x64) + b : 4 N-tiles each, K=128 ---
    for (int et = 0; et < 4; ++et) {
      v8f z0 = {};
      v8f a = wmma_gen<128>(s_q, 128, 1, P.q_w + et * 16 * 128, 1, 128, z0);
      tile_store(s_qq + et * 16, 64, a, 1.f, P.q_b + et * 16, nullptr, false, false);
      v8f b2 = wmma_gen<128>(s_cc, 128, 1, P.k_w + et * 16 * 128, 1, 128, z0);
      tile_store(s_kk + et * 16, 64, b2, 1.f, P.k_b + et * 16, nullptr, false, false);
      v8f d2 = wmma_gen<128>(s_cc, 128, 1, P.v_w + et * 16 * 128, 1, 128, z0);
      tile_store(s_vv + et * 16, 64, d2, 1.f, P.v_b + et * 16, nullptr, false, false);
    }
    __syncthreads();

    // --- aw[e,f] = 0.25 * sum_c qq[c,e]*kk[c,f]  (A = qq^T via strides) ---
    for (int et = 0; et < 4; ++et)
      for (int ft = 0; ft < 4; ++ft) {
        v8f z0 = {};
        v8f a = wmma_gen<16>(s_qq + et * 16, 1, 64, s_kk + ft * 16, 64, 1, z0);
        tile_store(s_aw + (et * 16) * 64 + ft * 16, 64, a, 0.25f, nullptr, nullptr,
                   false, false);
      }
    __syncthreads();

    // --- softmax over f (64 rows) ---
    for (int r2 = lane; r2 < 64; r2 += 32) {
      float mx = -3.0e38f;
      for (int f2 = 0; f2 < 64; ++f2) mx = fmaxf(mx, s_aw[r2 * 64 + f2]);
      float sm = 0.f;
      for (int f2 = 0; f2 < 64; ++f2) {
        float e2 = __expf(s_aw[r2 * 64 + f2] - mx);
        s_aw[r2 * 64 + f2] = e2; sm += e2;
      }
      float inv = 1.0f / sm;
      for (int f2 = 0; f2 < 64; ++f2) s_aw[r2 * 64 + f2] *= inv;
    }
    __syncthreads();

    // --- at[c,e] = sum_f vv[c,f]*aw[e,f] ---
    for (int et = 0; et < 4; ++et) {
      v8f z0 = {};
      v8f a = wmma_gen<64>(s_vv, 64, 1, s_aw + et * 16 * 64, 1, 64, z0);
      tile_store(s_at + et * 16, 64, a, 1.f, nullptr, nullptr, false, false);
    }
    __syncthreads();

    // --- at2[c,w] = sum_e at[c,e]*o_w[w,e] + o_b[w] ---
    for (int wt = 0; wt < 8; ++wt) {
      v8f z0 = {};
      v8f a = wmma_gen<64>(s_at, 64, 1, P.o_w + wt * 16 * 64, 1, 64, z0);
      tile_store(s_at2 + wt * 16, 128, a, 1.f, P.o_b + wt * 16, nullptr, false, false);
    }
    __syncthreads();

    // --- t4 = pin + fc3@at2 (4x128), then h = se_conv(t4) + b (16x128) ---
    float* tmp4 = s_big + 3072;
#pragma unroll
    for (int t = 0; t < 16; ++t) {
      int idx = lane + 32 * t;
      int o = idx >> 7, w = idx & 127;
      float a1 = rpin[t];
      for (int k = 0; k < 16; ++k) a1 += P.fc3_w[o * 16 + k] * s_at2[k * 128 + w];
      tmp4[idx] = a1;
    }
    __syncthreads();
    for (int idx = lane; idx < 2048; idx += 32) {
      int o = idx >> 7, w = idx & 127;
      float a1 = P.se_conv_b[o];
#pragma unroll
      for (int k = 0; k < 4; ++k) a1 += P.se_conv_w[o * 4 + k] * tmp4[k * 128 + w];
      s_h[idx] = a1;
    }
    __syncthreads();

    // --- SE block ---
    if (lane < 16) {
      float sm = 0.f;
      for (int w = 0; w < 128; ++w) sm += s_h[lane * 128 + w];
      s_stat[lane] = sm * (1.0f / 128.0f);
    }
    __syncthreads();
    if (lane < 8) {
      float a1 = P.se1_b[lane];
      for (int k = 0; k < 16; ++k) a1 += P.se1_w[lane * 16 + k] * s_stat[k];
      s_stat[16 + lane] = fmaxf(a1, 0.f);
    }
    __syncthreads();
    if (lane < 16) {
      float a1 = P.se2_b[lane];
      for (int k = 0; k < 8; ++k) a1 += P.se2_w[lane * 8 + k] * s_stat[16 + k];
      s_stat[lane] = 1.0f / (1.0f + __expf(-a1));
    }
    __syncthreads();
    for (int idx = lane; idx < 2048; idx += 32) s_h[idx] *= s_stat[idx >> 7];
    __syncthreads();

    // --- 3x gMLP ---
    for (int l = 0; l < 3; ++l) {
      const GmlpP& G = P.gl[l];
      if (lane < 16) {
        float m2 = 0.f;
        for (int w = 0; w < 128; ++w) m2 += s_h[lane * 128 + w];
        m2 *= (1.0f / 128.0f);
        float v2 = 0.f;
        for (int w = 0; w < 128; ++w) {
          float d3 = s_h[lane * 128 + w] - m2; v2 += d3 * d3;
        }
        v2 *= (1.0f / 128.0f);
        s_stat[lane] = m2;
        s_stat[16 + lane] = rsqrtf(v2 + 1e-5f);
      }
      __syncthreads();
      for (int idx = lane; idx < 2048; idx += 32) {
        int o = idx >> 7, w = idx & 127;
        s_hn[idx] = (s_h[idx] - s_stat[o]) * s_stat[16 + o] * G.ln_g[w] + G.ln_b[w];
      }
      __syncthreads();
      // z = gelu(f1 @ hn + f1_b)   (64x128)
      for (int et = 0; et < 4; ++et)
        for (int wt = 0; wt < 8; ++wt) {
          v8f z0 = {};
          v8f a = wmma_gen<16>(G.f1_w + et * 16 * 16, 16, 1, s_hn + wt * 16, 128, 1, z0);
          tile_store(s_z + et * 16 * 128 + wt * 16, 128, a, 1.f, nullptr,
                     G.f1_b + et * 16, true, false);
        }
      __syncthreads();
      // vp = BN_SCALE*(v @ proj^T) + proj_b   (v = z rows 32..63)
      for (int ct = 0; ct < 2; ++ct)
        for (int vt = 0; vt < 8; ++vt) {
          v8f z0 = {};
          v8f a = wmma_gen<128>(s_z + (32 + ct * 16) * 128, 128, 1,
                                G.proj_w + vt * 16 * 128, 1, 128, z0);
          tile_store(s_vp + ct * 16 * 128 + vt * 16, 128, a, 0.99999500f,
                     G.proj_b + vt * 16, nullptr, false, false);
        }
      __syncthreads();
      for (int idx = lane; idx < 4096; idx += 32) s_vp[idx] *= s_z[idx]; // uv = u*vp
      __syncthreads();
      // h += f2 @ uv + f2_b
      for (int wt = 0; wt < 8; ++wt) {
        v8f z0 = {};
        v8f a = wmma_gen<32>(G.f2_w, 32, 1, s_vp + wt * 16, 128, 1, z0);
        tile_store(s_h + wt * 16, 128, a, 1.f, nullptr, G.f2_b, false, true);
      }
      __syncthreads();
    }

    // --- head tail: pointwise transform, resize 128->513, write ang/mag, zsum ---
    for (int idx = lane; idx < 2048; idx += 32) {
      float v2 = s_h[idx];
      float tv;
      if (hs == 0) { float sg = 1.0f / (1.0f + __expf(-v2)); tv = sg * sg; }
      else         { float sp = softplusf(v2);               tv = sp * sp; }
      s_t[idx] = tv;
    }
    __syncthreads();
#pragma unroll
    for (int j = 0; j < 17; ++j) { zacc_r[j] = 0.f; zacc_i[j] = 0.f; }
    const float ratio = 128.0f / 513.0f;
    for (int ch = 0; ch < 16; ++ch) {
      float wc = 0.f;
      if (hs == 1) wc = softplusf(projw[ch]);
#pragma unroll
      for (int j = 0; j < 17; ++j) {
        int i2 = lane + 32 * j;
        if (i2 < 513) {
          float src = ((float)i2 + 0.5f) * ratio - 0.5f;
          float fs = floorf(src);
          int i0 = (int)fs;
          float fr = src - fs;
          int ia = min(max(i0, 0), 127), ib2 = min(max(i0 + 1, 0), 127);
          float yv = (1.0f - fr) * s_t[ch * 128 + ia] + fr * s_t[ch * 128 + ib2];
          size_t off = ((size_t)n * 16 + ch) * 513 + i2;
          if (hs == 0) {
            outAng[off] = gval + 12.0f * yv;            // (SUP_MS-INF_MS)=12, INF_MS=0
          } else {
            float ang = outAng[off];
            float ca = fmaxf(ang, 1.0f);
            float mg = yv / (ca * ca);
            outMag[off] = mg;
            float omg = 48.0f * ang * (-3.14159265358979f) * (float)i2 * (1.0f / 512.0f);
            float sn, cs2; __sincosf(omg, &sn, &cs2);
            zacc_r[j] += wc * mg * cs2;
            zacc_i[j] += wc * mg * sn;
          }
        }
      }
    }
    if (hs == 1) {
#pragma unroll
      for (int j = 0; j < 17; ++j) {
        int i2 = lane + 32 * j;
        if (i2 < 513) {
          zws[((size_t)n * 513 + i2) * 2]     = zacc_r[j];
          zws[((size_t)n * 513 + i2) * 2 + 1] = zacc_i[j];
        }
      }
    }
    __syncthreads();
  }
}

// ---------------------------------------------------------------------------
// Kernel B: out[n] = irfft( rfft(x[n],1024) * zsum[n], 1024 )
// Real-1024 via complex-512 radix-2 FFT in LDS; 256 threads = 1 butterfly each.
// ---------------------------------------------------------------------------
__device__ inline int brev9(int x) { return (int)(__brev((unsigned)x) >> 23); }

__device__ void fft512(float* re, float* im, float sign) {
  const int tid = threadIdx.x;
  for (int idx = tid; idx < 512; idx += 256) {
    int j = brev9(idx);
    if (j > idx) {
      float tr = re[idx]; re[idx] = re[j]; re[j] = tr;
      float ti = im[idx]; im[idx] = im[j]; im[j] = ti;
    }
  }
  __syncthreads();
  for (int len = 2; len <= 512; len <<= 1) {
    int half = len >> 1;
    int j = tid & (half - 1);
    int blk = tid / half;
    int i0 = blk * len + j;
    int i1 = i0 + half;
    float ang = sign * 6.28318530717958648f * (float)j / (float)len;
    float s, cc; __sincosf(ang, &s, &cc);
    float tr = cc * re[i1] - s * im[i1];
    float ti = cc * im[i1] + s * re[i1];
    re[i1] = re[i0] - tr; im[i1] = im[i0] - ti;
    re[i0] += tr;         im[i0] += ti;
    __syncthreads();
  }
}

__global__ __launch_bounds__(256) void NFS_fft_kernel(const float* __restrict__ x,
                                                      const float* __restrict__ zws,
                                                      float* __restrict__ out) {
  __shared__ float re[512], im[512], sXr[513], sXi[513];
  const int n = blockIdx.x;
  const int tid = threadIdx.x;
  const float* xg = x + (size_t)n * 1024;

  for (int m = tid; m < 512; m += 256) { re[m] = xg[2 * m]; im[m] = xg[2 * m + 1]; }
  __syncthreads();
  fft512(re, im, -1.0f);

  // rfft post-twiddle + multiply by zsum
  for (int k = tid; k <= 512; k += 256) {
    int k1 = k & 511, k2 = (512 - k) & 511;
    float vr1 = re[k1], vi1 = im[k1], vr2 = re[k2], vi2 = im[k2];
    float Per = 0.5f * (vr1 + vr2), Pei = 0.5f * (vi1 - vi2);
    float Mr = 0.5f * (vr1 - vr2), Mi = 0.5f * (vi1 + vi2);
    float Xor = Mi, Xoi = -Mr;                       // (V - conj(V2)) / (2i)
    float th = -3.14159265358979f * (float)k * (1.0f / 512.0f);
    float sw, cw; __sincosf(th, &sw, &cw);
    float Xr2 = Per + cw * Xor - sw * Xoi;
    float Xi2 = Pei + cw * Xoi + sw * Xor;
    float zr = zws[((size_t)n * 513 + k) * 2];
    float zi = zws[((size_t)n * 513 + k) * 2 + 1];
    sXr[k] = Xr2 * zr - Xi2 * zi;
    sXi[k] = Xr2 * zi + Xi2 * zr;
  }
  __syncthreads();

  // irfft pre-twiddle: V'[k] = Xe + i*Xo'
  for (int k = tid; k < 512; k += 256) {
    int k2 = 512 - k;
    float Yr1 = sXr[k], Yi1 = sXi[k], Yr2 = sXr[k2], Yi2 = sXi[k2];
    float Xer = 0.5f * (Yr1 + Yr2), Xei = 0.5f * (Yi1 - Yi2);
    float Mr = 0.5f * (Yr1 - Yr2), Mi = 0.5f * (Yi1 + Yi2);
    float th = 3.14159265358979f * (float)k * (1.0f / 512.0f);
    float sw, cw; __sincosf(th, &sw, &cw);
    float Xor = cw * Mr - sw * Mi;
    float Xoi = cw * Mi + sw * Mr;
    re[k] = Xer - Xoi;
    im[k] = Xei + Xor;
  }
  __syncthreads();
  fft512(re, im, 1.0f);

  float* og = out + (size_t)n * 1024;
  for (int m = tid; m < 512; m += 256) {
    og[2 * m]     = re[m] * (1.0f / 512.0f);
    og[2 * m + 1] = im[m] * (1.0f / 512.0f);
  }
}

// ---------------------------------------------------------------------------
// Host side
// ---------------------------------------------------------------------------
static EstP unpack_params(void* const* d, int base) {
  EstP p; int i = base;
  auto f = [&](void) { return (const float*)d[i++]; };
  // jax pytree order: sorted dict keys, gmlp list (each dict sorted) after fc3_w
  p.fc1_w = f(); p.fc2_w = f(); p.fc3_w = f();
  for (int l = 0; l < 3; ++l) {
    p.gl[l].f1_b = f(); p.gl[l].f1_w = f();
    p.gl[l].f2_b = f(); p.gl[l].f2_w = f();
    p.gl[l].ln_b = f(); p.gl[l].ln_g = f();
    p.gl[l].proj_b = f(); p.gl[l].proj_w = f();
  }
  p.k_b = f(); p.k_w = f();
  p.o_b = f(); p.o_w = f();
  p.q_b = f(); p.q_w = f();
  p.se1_b = f(); p.se1_w = f();
  p.se2_b = f(); p.se2_w = f();
  p.se_conv_b = f(); p.se_conv_w = f();
  p.v_b = f(); p.v_w = f();
  return p;
}

extern "C" void kernel_launch(void* const* d_in, const int* in_sizes, int n_in,
                              void* d_out, int out_size, void* d_ws, size_t ws_size,
                              hipStream_t stream) {
  const float* x = (const float*)d_in[0];
  const float* c = (const float*)d_in[1];
  const float* g = (const float*)d_in[2];
  // d_in[3] = b (unused scalar)
  EstP Psc = unpack_params(d_in, 4);        // params_scale: 41 leaves
  EstP Psh = unpack_params(d_in, 4 + 41);   // params_shift: 41 leaves
  const float* projw = (const float*)d_in[n_in - 1];

  const int N = in_sizes[2];                // 4096 (from g)
  float* out = (float*)d_out;               // (N,1,1024)
  float* outMag = out + (size_t)N * 1024;   // (N,16,513)
  float* outAng = outMag + (size_t)N * 16 * 513;
  float* zws = (float*)d_ws;                // N*513 complex (interleaved)

  NFS_est_kernel<<<N, 32, 0, stream>>>(c, g, Psh, Psc, projw, outMag, outAng, zws);
  NFS_fft_kernel<<<N, 256, 0, stream>>>(x, zws, out);
}